// Encoder_52475910423106
// MI455X (gfx1250) — compile-verified
//
#include <hip/hip_runtime.h>

#define N_NODES 100000
#define N_EDGES 1250000
#define C 64

typedef __attribute__((ext_vector_type(2))) float v2f;
typedef __attribute__((ext_vector_type(8))) float v8f;

// ---------------------------------------------------------------------------
// Zero-fill a float buffer (graph-capture-safe replacement for memset).
// ---------------------------------------------------------------------------
__global__ void zero_f32(float* __restrict__ p, long n) {
    long i = (long)blockIdx.x * blockDim.x + threadIdx.x;
    if (i < n) p[i] = 0.0f;
}

// ---------------------------------------------------------------------------
// Degree count: one thread per edge, hardware fp32 atomic (non-returning ->
// global_atomic_add_f32, STOREcnt, fire-and-forget).
// ---------------------------------------------------------------------------
__global__ void count_deg(const long long* __restrict__ dst,
                          float* __restrict__ cnt) {
    int e = blockIdx.x * blockDim.x + threadIdx.x;
    if (e < N_EDGES) {
        int d = (int)dst[e];
        unsafeAtomicAdd(&cnt[d], 1.0f);
    }
}

// ---------------------------------------------------------------------------
// Scatter-accumulate: 16 threads per edge, float4 per thread.
// One global_load_b128 gather + 4 non-returning fp32 atomics per thread.
// Whole working set (x, agg, edge_index ~ 71MB) is L2-resident (192MB), so
// this runs at L2 atomic throughput; wider requests cut the request count.
// ---------------------------------------------------------------------------
__global__ void scatter_accum(const long long* __restrict__ src,
                              const long long* __restrict__ dst,
                              const float* __restrict__ x,
                              float* __restrict__ agg) {
    long idx = (long)blockIdx.x * blockDim.x + threadIdx.x;
    if (idx >= (long)N_EDGES * (C / 4)) return;
    int e  = (int)(idx >> 4);        // edge
    int c4 = (int)(idx & 15) << 2;   // channel group base
    int s = (int)src[e];
    int d = (int)dst[e];
    float4 v = *(const float4*)(x + (long)s * C + c4);
    float* ap = agg + (long)d * C + c4;
    unsafeAtomicAdd(ap + 0, v.x);
    unsafeAtomicAdd(ap + 1, v.y);
    unsafeAtomicAdd(ap + 2, v.z);
    unsafeAtomicAdd(ap + 3, v.w);
}

// ---------------------------------------------------------------------------
// Fused SAGE node update with fp32 WMMA:
//   out = PReLU( (agg/max(cnt,1)) @ Wl + bl + x @ Wr , a )
// Block = 128 threads (4 waves). Block handles 16 rows; wave w handles the
// 16-column tile starting at 16*w. K=64 is consumed as 16 chained
// V_WMMA_F32_16X16X4_F32 ops per weight matrix (full fp32 precision).
//
// Fragment layouts per CDNA5 ISA 7.12.2:
//   A 16x4 f32 : lanes 0-15  -> M=lane,    v0:K=k0,   v1:K=k0+1
//                lanes 16-31 -> M=lane-16, v0:K=k0+2, v1:K=k0+3
//   B 4x16 f32 : symmetric with N striped across lanes
//   C/D 16x16  : v8f; lane 0-15 N=lane (M = v), lane 16-31 N=lane-16 (M = v+8)
// ---------------------------------------------------------------------------
__global__ __launch_bounds__(128) void sage_gemm_prelu(
    const float* __restrict__ agg, const float* __restrict__ cnt,
    const float* __restrict__ xin,
    const float* __restrict__ wl, const float* __restrict__ bl,
    const float* __restrict__ wr, const float* __restrict__ pa,
    float* __restrict__ out) {

    const int lane = threadIdx.x & 31;
    const int wave = threadIdx.x >> 5;
    const int row0 = blockIdx.x << 4;   // 16 rows per block, exact (100000/16)
    const int nb   = wave << 4;         // column tile base
    const int m    = lane & 15;         // row (A) / col (B,C) within tile
    const int kh   = (lane >> 4) << 1;  // K sub-offset: 0 or 2

    const float* arow = agg + (long)(row0 + m) * C;
    const float* xrow = xin + (long)(row0 + m) * C;
    // count is integer-valued; single v_rcp_f32 (~1 ulp) instead of precise div
    const float inv = __builtin_amdgcn_rcpf(fmaxf(cnt[row0 + m], 1.0f));

    v8f acc = {};

    // mean(agg) @ Wl
#pragma unroll
    for (int k0 = 0; k0 < C; k0 += 4) {
        v2f a, b;
        a.x = arow[k0 + kh] * inv;
        a.y = arow[k0 + kh + 1] * inv;
        b.x = wl[(k0 + kh) * C + nb + m];
        b.y = wl[(k0 + kh + 1) * C + nb + m];
        acc = __builtin_amdgcn_wmma_f32_16x16x4_f32(false, a, false, b,
                                                    (short)0, acc, false, false);
    }
    // x @ Wr (accumulate into same C)
#pragma unroll
    for (int k0 = 0; k0 < C; k0 += 4) {
        v2f a, b;
        a.x = xrow[k0 + kh];
        a.y = xrow[k0 + kh + 1];
        b.x = wr[(k0 + kh) * C + nb + m];
        b.y = wr[(k0 + kh + 1) * C + nb + m];
        acc = __builtin_amdgcn_wmma_f32_16x16x4_f32(false, a, false, b,
                                                    (short)0, acc, false, false);
    }

    // bias + channel-wise PReLU + store (C/D layout: N=lane&15, M=v + 8*(lane>=16))
    const int   col   = nb + m;
    const float bias  = bl[col];
    const float slope = pa[col];
    const int   rbase = (lane >> 4) << 3;  // 0 or 8
#pragma unroll
    for (int v = 0; v < 8; ++v) {
        float h = acc[v] + bias;
        h = (h >= 0.0f) ? h : slope * h;
        out[(long)(row0 + rbase + v) * C + col] = h;
    }
}

// ---------------------------------------------------------------------------
extern "C" void kernel_launch(void* const* d_in, const int* in_sizes, int n_in,
                              void* d_out, int out_size, void* d_ws, size_t ws_size,
                              hipStream_t stream) {
    (void)in_sizes; (void)n_in; (void)out_size; (void)ws_size;

    const float*     x   = (const float*)d_in[0];
    const long long* ei  = (const long long*)d_in[1];
    const long long* src = ei;
    const long long* dst = ei + N_EDGES;
    const float* wl0 = (const float*)d_in[2];
    const float* bl0 = (const float*)d_in[3];
    const float* wr0 = (const float*)d_in[4];
    const float* a0  = (const float*)d_in[5];
    const float* wl1 = (const float*)d_in[6];
    const float* bl1 = (const float*)d_in[7];
    const float* wr1 = (const float*)d_in[8];
    const float* a1  = (const float*)d_in[9];
    float* out = (float*)d_out;

    // workspace layout
    char* ws = (char*)d_ws;
    size_t cntBytes = ((size_t)N_NODES * 4 + 255) & ~(size_t)255;       // 400 KB
    size_t aggBytes = ((size_t)N_NODES * C * 4 + 255) & ~(size_t)255;   // 25.6 MB
    float* cnt = (float*)ws;
    float* agg = (float*)(ws + cntBytes);
    float* h1  = (float*)(ws + cntBytes + aggBytes);

    const dim3 b256(256);
    const long aggN   = (long)N_NODES * C;            // 6.4M
    const long scatN  = (long)N_EDGES * (C / 4);      // 20M scatter threads
    const int  gNodes = (N_NODES + 255) / 256;
    const int  gEdges = (N_EDGES + 255) / 256;
    const int  gAgg   = (int)((aggN + 255) / 256);
    const int  gScat  = (int)((scatN + 255) / 256);
    const int  gGemm  = N_NODES / 16;                 // exact: 6250

    // degrees (shared by both layers)
    zero_f32<<<gNodes, b256, 0, stream>>>(cnt, (long)N_NODES);
    count_deg<<<gEdges, b256, 0, stream>>>(dst, cnt);

    // layer 0: x -> h1
    zero_f32<<<gAgg, b256, 0, stream>>>(agg, aggN);
    scatter_accum<<<gScat, b256, 0, stream>>>(src, dst, x, agg);
    sage_gemm_prelu<<<gGemm, dim3(128), 0, stream>>>(agg, cnt, x,
                                                     wl0, bl0, wr0, a0, h1);

    // layer 1: h1 -> out
    zero_f32<<<gAgg, b256, 0, stream>>>(agg, aggN);
    scatter_accum<<<gScat, b256, 0, stream>>>(src, dst, h1, agg);
    sage_gemm_prelu<<<gGemm, dim3(128), 0, stream>>>(agg, cnt, h1,
                                                     wl1, bl1, wr1, a1, out);
}